// GraphConvLayer_15126874817099
// MI455X (gfx1250) — compile-verified
//
#include <hip/hip_runtime.h>

// GCN layer for MI455X (gfx1250, wave32).
//   d_in[0]=x[N,64] f32, d_in[1]=edge_index[2,E] i32, d_in[2]=x0 (unused),
//   d_in[3]=W[64,64] f32 (row-major, W[o][d]), d_in[4]=bias[64] f32
//   d_out = agg buffer then final [N,64] f32 (in-place WMMA epilogue).

#define DCH 64

typedef __attribute__((ext_vector_type(2))) float v2f;
typedef __attribute__((ext_vector_type(8))) float v8f;

// ---- Pass 1: deg[n] = 1 (self loop) -----------------------------------------
__global__ void k_init_deg(float* __restrict__ deg, int N) {
    int i = blockIdx.x * blockDim.x + threadIdx.x;
    if (i < N) deg[i] = 1.0f;
}

// ---- Pass 2: deg[col[e]] += 1 ----------------------------------------------
__global__ void k_count_deg(const int* __restrict__ col, float* __restrict__ deg, int E) {
    int e = blockIdx.x * blockDim.x + threadIdx.x;
    if (e < E) atomicAdd(&deg[col[e]], 1.0f);
}

// ---- Pass 3: dinv[n]=rsqrt(deg); agg[n][:] = dinv^2 * x[n][:] (self loop) ---
__global__ void k_init_agg(const float* __restrict__ x, const float* __restrict__ deg,
                           float* __restrict__ dinv, float* agg, int total) {
    int idx = blockIdx.x * blockDim.x + threadIdx.x;
    if (idx >= total) return;
    int n = idx >> 6;                 // D = 64
    float di = rsqrtf(deg[n]);        // deg >= 1, always valid
    if ((idx & (DCH - 1)) == 0) dinv[n] = di;
    agg[idx] = di * di * x[idx];
}

// ---- Pass 4: SpMM scatter. One wave32 per edge, float2 per lane (64 ch). ----
__global__ void k_scatter(const int* __restrict__ row, const int* __restrict__ col,
                          const float* __restrict__ dinv, const float* __restrict__ x,
                          float* agg, int E) {
    int gid  = blockIdx.x * blockDim.x + threadIdx.x;
    int e    = gid >> 5;
    int lane = gid & 31;
    if (e >= E) return;
    int r = row[e], c = col[e];
    float nrm = dinv[r] * dinv[c];
    float2 v = ((const float2*)(x + (size_t)r * DCH))[lane];   // coalesced 256B gather
    float* dst = agg + (size_t)c * DCH + lane * 2;
    atomicAdd(dst + 0, nrm * v.x);                             // global_atomic_add_f32
    atomicAdd(dst + 1, nrm * v.y);
}

// ---- Pass 5: out = agg @ W^T + bias via V_WMMA_F32_16X16X4_F32 --------------
// One wave per 16-row slab, 4 N-tiles of 16 cols, K=64 in steps of 4.
// A fragment (16x4): lane L -> M=L&15, K pair kk+2*(L>=16) (contiguous float2).
// B fragment (4x16): B[k][n] = W[n][k]  -> contiguous float2 from W row n.
// C/D: VGPR r -> row r+8*(L>=16), col L&15.
__global__ void k_gemm_wmma(const float* agg, const float* __restrict__ W,
                            const float* __restrict__ bias, float* out, int N) {
    int wave = (blockIdx.x * blockDim.x + threadIdx.x) >> 5;
    int lane = threadIdx.x & 31;
    int m0 = wave * 16;
    if (m0 >= N) return;                  // whole wave exits: EXEC stays all-ones
    int hi = lane >> 4;                   // 0|1 : selects K pair / output row half
    int lo = lane & 15;

    v8f acc[4];
#pragma unroll
    for (int t = 0; t < 4; ++t) {
        float bv = bias[t * 16 + lo];     // bias depends on N (col) only
#pragma unroll
        for (int r = 0; r < 8; ++r) acc[t][r] = bv;
    }

    const float* arow  = agg + (size_t)(m0 + lo) * DCH + 2 * hi;
    const float* wbase = W + 2 * hi;

#pragma unroll
    for (int kk = 0; kk < DCH; kk += 4) {
        v2f a = *(const v2f*)(arow + kk);
#pragma unroll
        for (int t = 0; t < 4; ++t) {
            v2f b = *(const v2f*)(wbase + (size_t)(t * 16 + lo) * DCH + kk);
            acc[t] = __builtin_amdgcn_wmma_f32_16x16x4_f32(
                false, a, false, b, (short)0, acc[t], false, false);
        }
    }

    float* orow = out + (size_t)m0 * DCH;
#pragma unroll
    for (int t = 0; t < 4; ++t)
#pragma unroll
        for (int r = 0; r < 8; ++r)
            orow[(size_t)(r + 8 * hi) * DCH + t * 16 + lo] = acc[t][r];
}

extern "C" void kernel_launch(void* const* d_in, const int* in_sizes, int n_in,
                              void* d_out, int out_size, void* d_ws, size_t ws_size,
                              hipStream_t stream) {
    const float* x    = (const float*)d_in[0];
    const int*   ei   = (const int*)d_in[1];
    const float* W    = (const float*)d_in[3];
    const float* bias = (const float*)d_in[4];

    int N = in_sizes[0] / DCH;
    int E = in_sizes[1] / 2;
    const int* row = ei;          // edge_index[0] = source nodes
    const int* col = ei + E;      // edge_index[1] = target nodes

    float* deg  = (float*)d_ws;   // [N]
    float* dinv = deg + N;        // [N]   (only 2*N*4 bytes of ws used)
    float* agg  = (float*)d_out;  // [N,64] aggregated in-place, then GEMM epilogue
    float* out  = (float*)d_out;

    int total = N * DCH;
    k_init_deg <<<(N + 255) / 256, 256, 0, stream>>>(deg, N);
    k_count_deg<<<(E + 255) / 256, 256, 0, stream>>>(col, deg, E);
    k_init_agg <<<(total + 255) / 256, 256, 0, stream>>>(x, deg, dinv, agg, total);

    long long sthreads = (long long)E * 32;
    k_scatter  <<<(int)((sthreads + 255) / 256), 256, 0, stream>>>(row, col, dinv, x, agg, E);

    int mwaves = (N + 15) / 16;   // N=100000 -> 6250 slabs, 8 waves/block
    k_gemm_wmma<<<(mwaves * 32 + 255) / 256, 256, 0, stream>>>(agg, W, bias, out, N);
}